// HierarchicalGNN_59631325937740
// MI455X (gfx1250) — compile-verified
//
#include <hip/hip_runtime.h>
#include <hip/hip_bf16.h>

// ---------------------------------------------------------------------------
// HierarchicalGNN for MI455X (gfx1250, wave32, WMMA)
//   N_ATOM=131072, E_ATOM=524288, N_FRAG=32768, E_FRAG=131072, HID=256, FEAT=9
// ---------------------------------------------------------------------------

#define N_ATOM 131072
#define E_ATOM 524288
#define N_FRAG 32768
#define E_FRAG 131072
#define HID    256
#define FEAT   9

typedef __attribute__((ext_vector_type(2))) float v2f;
typedef __attribute__((ext_vector_type(8))) float v8f;

// ---------------------------------------------------------------------------
// Input projection: Y[i,h] = sum_k X[i,k] * W[k,h] + b[h] (+ add[i,h])
// One block per row, 256 threads (HID).
// ---------------------------------------------------------------------------
__global__ void proj9_kernel(const float* __restrict__ X,
                             const float* __restrict__ W,
                             const float* __restrict__ b,
                             const float* __restrict__ add,  // nullable
                             float* __restrict__ Y) {
  __shared__ float xr[FEAT];
  const int row = blockIdx.x;
  const int h = threadIdx.x;
  if (h < FEAT) xr[h] = X[(size_t)row * FEAT + h];
  __syncthreads();
  float acc = b[h];
#pragma unroll
  for (int k = 0; k < FEAT; ++k) acc = fmaf(xr[k], W[k * HID + h], acc);
  if (add) acc += add[(size_t)row * HID + h];
  Y[(size_t)row * HID + h] = acc;
}

// ---------------------------------------------------------------------------
// fill
// ---------------------------------------------------------------------------
__global__ void fill_kernel(float* __restrict__ p, int n, float v) {
  int i = blockIdx.x * blockDim.x + threadIdx.x;
  if (i < n) p[i] = v;
}

// ---------------------------------------------------------------------------
// Edge prep: ew[e] = 1 - sigmoid(raw[e]); deg[dst[e]] += ew[e]
// ---------------------------------------------------------------------------
__global__ void edge_prep_kernel(const float* __restrict__ raw,
                                 const int* __restrict__ eidx, int E,
                                 float* __restrict__ ew,
                                 float* __restrict__ deg) {
  int e = blockIdx.x * blockDim.x + threadIdx.x;
  if (e >= E) return;
  float s = 1.0f / (1.0f + __expf(-raw[e]));
  float w = 1.0f - s;
  ew[e] = w;
  atomicAdd(&deg[eidx[E + e]], w);
}

// deg -> deg^-0.5 (in place)
__global__ void dinv_kernel(float* __restrict__ deg, int n) {
  int i = blockIdx.x * blockDim.x + threadIdx.x;
  if (i < n) {
    float d = deg[i];
    deg[i] = (d > 0.0f) ? rsqrtf(d) : 0.0f;
  }
}

// ---------------------------------------------------------------------------
// agg[i,:] = dinv[i]^2 * H[i,:]   (self-loop term; full init of agg)
// ---------------------------------------------------------------------------
__global__ void self_init_kernel(const float* __restrict__ H,
                                 const float* __restrict__ dinv,
                                 float* __restrict__ agg) {
  const int i = blockIdx.x;
  const int h = threadIdx.x;
  float di = dinv[i];
  agg[(size_t)i * HID + h] = di * di * H[(size_t)i * HID + h];
}

// ---------------------------------------------------------------------------
// agg[dst,:] += dinv[src]*ew[e]*dinv[dst] * H[src,:]   (one block per edge)
// ---------------------------------------------------------------------------
__global__ void edge_scatter_kernel(const float* __restrict__ H,
                                    const int* __restrict__ eidx, int E,
                                    const float* __restrict__ ew,
                                    const float* __restrict__ dinv,
                                    float* __restrict__ agg) {
  const int e = blockIdx.x;
  const int src = eidx[e];
  const int dst = eidx[E + e];
  const float coef = dinv[src] * ew[e] * dinv[dst];
  const int h = threadIdx.x;
  atomicAdd(&agg[(size_t)dst * HID + h], coef * H[(size_t)src * HID + h]);
}

// y = relu(agg + b)
__global__ void relu_bias_kernel(const float* __restrict__ agg,
                                 const float* __restrict__ b,
                                 float* __restrict__ y) {
  const int i = blockIdx.x;
  const int h = threadIdx.x;
  y[(size_t)i * HID + h] = fmaxf(agg[(size_t)i * HID + h] + b[h], 0.0f);
}

// ---------------------------------------------------------------------------
// Repack W [256x256] row-major into WMMA B-fragment order:
//   Wp[((kstep*16 + t)*32 + lane)*2 + {0,1}] = W[kstep*4+kb(+1)][t*16 + (lane&15)]
//   with kb = (lane>>4)*2   (B 4x16 layout: v0 = rows K0/K2, v1 = rows K1/K3)
// ---------------------------------------------------------------------------
__global__ void repack_w_kernel(const float* __restrict__ W,
                                float* __restrict__ Wp) {
  int idx = blockIdx.x * blockDim.x + threadIdx.x;  // 64*16*32 = 32768
  int lane = idx & 31;
  int t = (idx >> 5) & 15;
  int kstep = idx >> 9;
  int kb = (lane >> 4) << 1;
  int col = t * 16 + (lane & 15);
  int row = kstep * 4 + kb;
  Wp[(size_t)idx * 2 + 0] = W[row * HID + col];
  Wp[(size_t)idx * 2 + 1] = W[(row + 1) * HID + col];
}

// ---------------------------------------------------------------------------
// GEMM: Y[N,256] = X[N,256] @ W[256,256] (+bias[col]) (*rowscale[row])
// via V_WMMA_F32_16X16X4_F32. Block = 64 threads (2 waves), 32 rows/block.
// X tile staged in LDS (padded stride 260 -> conflict-free fragment reads);
// B fragments come pre-packed (coalesced global_load_b64).
// ---------------------------------------------------------------------------
#define LDS_STRIDE 260

__global__ __launch_bounds__(64) void gemm256_wmma_kernel(
    const float* __restrict__ X, const float* __restrict__ Wp,
    const float* __restrict__ bias,      // nullable
    const float* __restrict__ rowscale,  // nullable
    float* __restrict__ Y, int N) {
  __shared__ float lds[32 * LDS_STRIDE];

  const int tid = threadIdx.x;
  const int rowBase = blockIdx.x * 32;

  // Prefetch next row block of X (speculative, streams X through L2).
  if (rowBase + 32 < N)
    __builtin_prefetch(X + (size_t)(rowBase + 32) * HID + tid * 4, 0, 0);

  // Stage 32x256 X tile into LDS (coalesced float4 loads).
  {
    const float4* Xv = (const float4*)(X + (size_t)rowBase * HID);
    for (int i = tid; i < 32 * 64; i += 64) {
      int r = i >> 6;            // 64 float4 per row
      int c = (i & 63) << 2;     // column in floats
      float4 v = Xv[i];
      *(float4*)&lds[r * LDS_STRIDE + c] = v;
    }
  }
  __syncthreads();

  const int lane = tid & 31;
  const int wv = tid >> 5;               // wave id 0..1 -> 16-row group
  const int m = lane & 15;               // A-fragment row within tile
  const int kb = (lane >> 4) << 1;       // K sub-offset 0 or 2

  v8f acc[16] = {};                      // 16 col tiles x 8 VGPRs

  const float* arow = &lds[(wv * 16 + m) * LDS_STRIDE];

  for (int k = 0; k < HID; k += 4) {
    v2f a;
    a.x = arow[k + kb];
    a.y = arow[k + kb + 1];
    const float* wp = Wp + ((size_t)(k >> 2) << 10) + (size_t)lane * 2;
#pragma unroll
    for (int t = 0; t < 16; ++t) {
      v2f b = *(const v2f*)(wp + t * 64);
      acc[t] = __builtin_amdgcn_wmma_f32_16x16x4_f32(
          false, a, false, b, (short)0, acc[t], false, false);
    }
  }

  // Epilogue: D layout -> row = rowBase + wv*16 + (lane>>4)*8 + i, col = t*16 + (lane&15)
  const int rb = rowBase + wv * 16 + ((lane >> 4) << 3);
  const int cb = lane & 15;
#pragma unroll
  for (int t = 0; t < 16; ++t) {
    const int col = t * 16 + cb;
    float badd = bias ? bias[col] : 0.0f;
#pragma unroll
    for (int i = 0; i < 8; ++i) {
      int row = rb + i;
      float v = acc[t][i] + badd;
      if (rowscale) v *= rowscale[row];
      Y[(size_t)row * HID + col] = v;
    }
  }
}

// ---------------------------------------------------------------------------
// Fragment pooling helpers
// ---------------------------------------------------------------------------
__global__ void cnt_kernel(const int* __restrict__ a2f, float* __restrict__ cnt) {
  int i = blockIdx.x * blockDim.x + threadIdx.x;
  if (i < N_ATOM) atomicAdd(&cnt[a2f[i]], 1.0f);
}

__global__ void fragsum_kernel(const float* __restrict__ P,
                               const int* __restrict__ a2f,
                               float* __restrict__ fsum) {
  const int i = blockIdx.x;
  const int f = a2f[i];
  const int h = threadIdx.x;
  atomicAdd(&fsum[(size_t)f * HID + h], P[(size_t)i * HID + h]);
}

__global__ void combine_frag_kernel(const float* __restrict__ xf0,
                                    const float* __restrict__ fsum,
                                    const float* __restrict__ cnt,
                                    float* __restrict__ xf1) {
  const int f = blockIdx.x;
  const int h = threadIdx.x;
  float c = fmaxf(cnt[f], 1.0f);
  xf1[(size_t)f * HID + h] =
      xf0[(size_t)f * HID + h] + fsum[(size_t)f * HID + h] / c;
}

// xa_out[i,:] += Q[a2f[i],:] * alpha[i]
__global__ void broadcast_add_kernel(float* __restrict__ xa,
                                     const float* __restrict__ Q,
                                     const int* __restrict__ a2f,
                                     const float* __restrict__ alpha) {
  const int i = blockIdx.x;
  const int f = a2f[i];
  const float al = alpha[i];
  const int h = threadIdx.x;
  xa[(size_t)i * HID + h] += Q[(size_t)f * HID + h] * al;
}

// ---------------------------------------------------------------------------
// Orchestration
// ---------------------------------------------------------------------------
extern "C" void kernel_launch(void* const* d_in, const int* in_sizes, int n_in,
                              void* d_out, int out_size, void* d_ws, size_t ws_size,
                              hipStream_t stream) {
  const float* x_atom   = (const float*)d_in[0];
  const int*   eidx_a   = (const int*)d_in[1];
  const float* frag_h   = (const float*)d_in[2];
  const int*   eidx_f   = (const int*)d_in[3];
  const int*   a2f      = (const int*)d_in[4];
  const float* s0       = (const float*)d_in[5];
  const float* alpha    = (const float*)d_in[6];
  const float* curv     = (const float*)d_in[7];
  const float* few_raw  = (const float*)d_in[8];
  const float* W1  = (const float*)d_in[9];
  const float* b1  = (const float*)d_in[10];
  const float* W2  = (const float*)d_in[11];
  const float* b2  = (const float*)d_in[12];
  const float* Wg1 = (const float*)d_in[13];
  const float* bg1 = (const float*)d_in[14];
  const float* Wg2 = (const float*)d_in[15];
  const float* bg2 = (const float*)d_in[16];
  const float* Wa2f = (const float*)d_in[17];
  const float* ba2f = (const float*)d_in[18];
  const float* Wf2a = (const float*)d_in[19];
  const float* bf2a = (const float*)d_in[20];

  float* out_xa = (float*)d_out;                              // [N_ATOM,256]
  float* out_xf = (float*)d_out + (size_t)N_ATOM * HID;       // [N_FRAG,256]

  // Workspace carve-up (floats).
  float* ws = (float*)d_ws;
  float* xa0  = ws;                                           // 33.5M
  float* H1   = xa0 + (size_t)N_ATOM * HID;                   // 33.5M (H/P/Q reuse)
  float* agg  = H1 + (size_t)N_ATOM * HID;                    // 33.5M
  float* xf0  = agg + (size_t)N_ATOM * HID;                   // 8.4M
  float* fsum = xf0 + (size_t)N_FRAG * HID;                   // 8.4M
  float* degA = fsum + (size_t)N_FRAG * HID;                  // 131072
  float* ewA  = degA + N_ATOM;                                // 524288
  float* degF = ewA + E_ATOM;                                 // 32768
  float* ewF  = degF + N_FRAG;                                // 131072
  float* cnt  = ewF + E_FRAG;                                 // 32768
  float* Wp   = cnt + N_FRAG;                                 // 65536

  // 1. Input projections.
  proj9_kernel<<<N_ATOM, HID, 0, stream>>>(x_atom, W1, b1, s0, xa0);
  proj9_kernel<<<N_FRAG, HID, 0, stream>>>(frag_h, W2, b2, nullptr, xf0);

  // 2. Atom-edge weights / degrees / d^-1/2 (self-loop -> deg init 1).
  fill_kernel<<<N_ATOM / 256, 256, 0, stream>>>(degA, N_ATOM, 1.0f);
  edge_prep_kernel<<<E_ATOM / 256, 256, 0, stream>>>(curv, eidx_a, E_ATOM, ewA, degA);
  dinv_kernel<<<N_ATOM / 256, 256, 0, stream>>>(degA, N_ATOM);

  // 3. Atom GCN: H1 = xa0 @ Wg1  (WMMA)
  repack_w_kernel<<<128, 256, 0, stream>>>(Wg1, Wp);
  gemm256_wmma_kernel<<<N_ATOM / 32, 64, 0, stream>>>(xa0, Wp, nullptr, nullptr,
                                                      H1, N_ATOM);
  // 4. Normalized scatter-add + self loop, then relu(+bg1) -> out_xa (== xa1)
  self_init_kernel<<<N_ATOM, HID, 0, stream>>>(H1, degA, agg);
  edge_scatter_kernel<<<E_ATOM, HID, 0, stream>>>(H1, eidx_a, E_ATOM, ewA, degA, agg);
  relu_bias_kernel<<<N_ATOM, HID, 0, stream>>>(agg, bg1, out_xa);

  // 5. P = (xa1 @ Wa2f + ba2f) * alpha  (WMMA, fused epilogue) -> reuse H1
  repack_w_kernel<<<128, 256, 0, stream>>>(Wa2f, Wp);
  gemm256_wmma_kernel<<<N_ATOM / 32, 64, 0, stream>>>(out_xa, Wp, ba2f, alpha,
                                                      H1, N_ATOM);

  // 6. Atom -> fragment mean pooling.
  fill_kernel<<<N_FRAG / 256, 256, 0, stream>>>(cnt, N_FRAG, 0.0f);
  fill_kernel<<<N_FRAG * HID / 256, 256, 0, stream>>>(fsum, N_FRAG * HID, 0.0f);
  cnt_kernel<<<N_ATOM / 256, 256, 0, stream>>>(a2f, cnt);
  fragsum_kernel<<<N_ATOM, HID, 0, stream>>>(H1, a2f, fsum);
  combine_frag_kernel<<<N_FRAG, HID, 0, stream>>>(xf0, fsum, cnt, xf0);

  // 7. Fragment GCN.
  fill_kernel<<<N_FRAG / 256, 256, 0, stream>>>(degF, N_FRAG, 1.0f);
  edge_prep_kernel<<<E_FRAG / 256, 256, 0, stream>>>(few_raw, eidx_f, E_FRAG, ewF, degF);
  dinv_kernel<<<N_FRAG / 256, 256, 0, stream>>>(degF, N_FRAG);

  repack_w_kernel<<<128, 256, 0, stream>>>(Wg2, Wp);
  gemm256_wmma_kernel<<<N_FRAG / 32, 64, 0, stream>>>(xf0, Wp, nullptr, nullptr,
                                                      H1, N_FRAG);
  self_init_kernel<<<N_FRAG, HID, 0, stream>>>(H1, degF, agg);
  edge_scatter_kernel<<<E_FRAG, HID, 0, stream>>>(H1, eidx_f, E_FRAG, ewF, degF, agg);
  relu_bias_kernel<<<N_FRAG, HID, 0, stream>>>(agg, bg2, out_xf);  // x_frag output

  // 8. Fragment -> atom broadcast: Q = x_frag @ Wf2a + bf2a; xa += Q[a2f]*alpha
  repack_w_kernel<<<128, 256, 0, stream>>>(Wf2a, Wp);
  gemm256_wmma_kernel<<<N_FRAG / 32, 64, 0, stream>>>(out_xf, Wp, bf2a, nullptr,
                                                      H1, N_FRAG);
  broadcast_add_kernel<<<N_ATOM, HID, 0, stream>>>(out_xa, H1, a2f, alpha);
}